// ScenePatch_64020782514345
// MI455X (gfx1250) — compile-verified
//
#include <hip/hip_runtime.h>
#include <hip/hip_bf16.h>

typedef __attribute__((ext_vector_type(2))) float v2f;
typedef __attribute__((ext_vector_type(8))) float v8f;

#define BATCH    32
#define IN_DIMS  128
#define PATCH    64
#define CHANNELS 3
#define AREA     (PATCH * PATCH * CHANNELS)   // 12288
#define CANVAS   512
#define IMG_PER_B (2 * AREA)                  // 24576 floats = 6 channels * 64*64
#define TILE     128                          // render tile edge (pixels)

// TDM arity dispatch (device pass only):
//   clang-22 (ROCm 7.2):        5-arg builtin (probe-confirmed)
//   clang-23 (amdgpu-toolchain): 6-arg builtin (probe-confirmed, zero-filled 5th)
//   builtin absent / host pass:  manual staging loop
#define TDM_ARITY 0
#if defined(__AMDGCN__) && defined(__has_builtin)
#if __has_builtin(__builtin_amdgcn_tensor_load_to_lds)
#undef TDM_ARITY
#if __clang_major__ >= 23
#define TDM_ARITY 6
#else
#define TDM_ARITY 5
#endif
#endif
#endif

#if TDM_ARITY
typedef __attribute__((ext_vector_type(4))) unsigned int u32x4;
typedef __attribute__((ext_vector_type(8))) int i32x8;
typedef __attribute__((ext_vector_type(4))) int i32x4;
#endif

// ---------------------------------------------------------------------------
// Kernel 1: theta[b][j] = b_z[b,:] . trans_w[j,:] + trans_b[j]   (32x6 values)
// ---------------------------------------------------------------------------
__global__ void k_theta(const float* __restrict__ b_z,
                        const float* __restrict__ trans_w,
                        const float* __restrict__ trans_b,
                        float* __restrict__ theta) {
    int t = blockIdx.x * blockDim.x + threadIdx.x;
    if (t >= BATCH * 6) return;
    int b = t / 6, j = t % 6;
    const float* zr = b_z + b * IN_DIMS;
    const float* wr = trans_w + j * IN_DIMS;
    float acc = trans_b[j];
#pragma unroll 8
    for (int k = 0; k < IN_DIMS; ++k) acc = fmaf(zr[k], wr[k], acc);
    theta[t] = acc;
}

// ---------------------------------------------------------------------------
// Kernel 2: WMMA f32 GEMMs (M=32, N=12288, K=128) for patch & alpha heads,
// fused activation epilogue; writes premultiplied image:
//   img[b][c]      = tanh(pre_p) * sigmoid(pre_a)   c = 0..2
//   img[b][3 + c]  = sigmoid(pre_a)                 c = 0..2
// One wave per 16x16 output tile. 2 M-tiles x 768 N-tiles = 1536 tiles.
//
// V_WMMA_F32_16X16X4_F32 operand layout (ISA 7.12.2, 32-bit 16x4 A-matrix):
//   A lane m (0-15): VGPR0=A[m,k0],   VGPR1=A[m,k0+1]
//   A lane 16+m   : VGPR0=A[m,k0+2], VGPR1=A[m,k0+3]   -> contiguous f32 pair
//   B mirrored with N striped over lanes; C/D: (M = r + 8*(lane>=16), N = lane&15)
// ---------------------------------------------------------------------------
__global__ __launch_bounds__(256) void k_gen(const float* __restrict__ b_z,
                                             const float* __restrict__ patch_w,
                                             const float* __restrict__ patch_b,
                                             const float* __restrict__ alpha_w,
                                             const float* __restrict__ alpha_b,
                                             float* __restrict__ img) {
    int lane = threadIdx.x & 31;
    int tile = blockIdx.x * (blockDim.x >> 5) + (threadIdx.x >> 5);
    int mt = tile & 1;        // 0..1   (batch tile)
    int nt = tile >> 1;       // 0..767 (area tile)
    int m0 = mt * 16;
    int n0 = nt * 16;
    int hi  = lane >> 4;      // 0: lanes 0-15, 1: lanes 16-31
    int lid = lane & 15;

    const float* arow = b_z     + (m0 + lid) * IN_DIMS;  // A row (batch m0+lid)
    const float* prow = patch_w + (n0 + lid) * IN_DIMS;  // B col n=lid: B[k,n]=W[n,k]
    const float* qrow = alpha_w + (n0 + lid) * IN_DIMS;
    int koff = hi * 2;

    v8f accP = {};
    v8f accA = {};
#pragma unroll 4
    for (int k0 = 0; k0 < IN_DIMS; k0 += 4) {
        v2f a  = { arow[k0 + koff], arow[k0 + koff + 1] };
        v2f bp = { prow[k0 + koff], prow[k0 + koff + 1] };
        v2f ba = { qrow[k0 + koff], qrow[k0 + koff + 1] };
        accP = __builtin_amdgcn_wmma_f32_16x16x4_f32(
                   false, a, false, bp, (short)0, accP, false, false);
        accA = __builtin_amdgcn_wmma_f32_16x16x4_f32(
                   false, a, false, ba, (short)0, accA, false, false);
    }

    int aidx = n0 + lid;                  // area index (c*4096 + y*64 + x)
    float pb = patch_b[aidx];
    float ab = alpha_b[aidx];
#pragma unroll
    for (int r = 0; r < 8; ++r) {
        int bb = m0 + hi * 8 + r;         // batch index for this C/D row
        float p  = tanhf(accP[r] + pb);
        float al = 1.0f / (1.0f + __expf(-(accA[r] + ab)));
        float* dst = img + (size_t)bb * IMG_PER_B;
        dst[aidx]        = p * al;        // premultiplied patch channels 0..2
        dst[AREA + aidx] = al;            // alpha channels 3..5
    }
}

// ---------------------------------------------------------------------------
// Kernel 3: affine grid + bilinear sample. One block per (batch, 128x128
// canvas tile); the 6x64x64 source image (96 KB) is staged into LDS — by the
// Tensor Data Mover (one TENSOR_LOAD_TO_LDS descriptor per block, TENSORcnt-
// tracked) when the builtin is available, else by a vectorized copy loop —
// and amortized over 16K output pixels. 64 pixels per thread; planar b32 LDS
// gathers are bank-conflict-free (adjacent lanes -> adjacent x). Output is
// write-once/never-read and bigger than the 192 MB L2, so stores are
// non-temporal to stream past L2 and keep img resident.
// ---------------------------------------------------------------------------
__global__ __launch_bounds__(256) void k_render(const float* __restrict__ img,
                                                const float* __restrict__ theta,
                                                float* __restrict__ y_out,
                                                float* __restrict__ m_out) {
    extern __shared__ float simg[];       // IMG_PER_B floats = 96 KB
    int b = blockIdx.z;
    const float* src = img + (size_t)b * IMG_PER_B;

#if TDM_ARITY
    if (threadIdx.x < 32) {               // one wave issues the block's DMA
        unsigned long long ga = (unsigned long long)(uintptr_t)src;
        unsigned int lds_off  = (unsigned int)(uintptr_t)simg; // aperture: addr[31:0]

        // D# group 0: count=1 | lds_addr | global_addr[56:0] | type=2
        u32x4 g0 = { 1u,
                     lds_off,
                     (unsigned int)ga,
                     (unsigned int)((ga >> 32) & 0x01FFFFFFu) | (2u << 30) };
        // D# group 1: 1-D copy of 24576 4-byte elements, no multicast/pad/iter
        i32x8 g1 = { (int)(2u << 16),                  // data_size=4B; wg_mask=0
                     (int)((unsigned)IMG_PER_B << 16), // tensor_dim0[15:0]<<16
                     0,                                // dim0[31:16]=0, dim1 lo=0
                     (int)((unsigned)IMG_PER_B << 16), // tile_dim0 = 24576
                     0,                                // tile_dim1/2 = 0 (unused)
                     IMG_PER_B,                        // tensor_dim0_stride[31:0]
                     0, 0 };
        i32x4 g2 = { 0, 0, 0, 0 };
        i32x4 g3 = { 0, 0, 0, 0 };
#if TDM_ARITY == 6
        i32x8 g4 = { 0, 0, 0, 0, 0, 0, 0, 0 };
        __builtin_amdgcn_tensor_load_to_lds(g0, g1, g2, g3, g4, 0);
#else
        __builtin_amdgcn_tensor_load_to_lds(g0, g1, g2, g3, 0);
#endif
        __builtin_amdgcn_s_wait_tensorcnt(0);
    }
#else
    for (int i = threadIdx.x * 4; i < IMG_PER_B; i += 256 * 4) {
        *(float4*)(simg + i) = *(const float4*)(src + i);
    }
#endif
    __syncthreads();

    const float* th = theta + b * 6;
    float t00 = th[0], t01 = th[1], t02 = th[2];
    float t10 = th[3], t11 = th[4], t12 = th[5];

    int tx0 = blockIdx.x * TILE;
    int ty0 = blockIdx.y * TILE;
    size_t obase = (size_t)b * CHANNELS * CANVAS * CANVAS;

#pragma unroll 2
    for (int i = 0; i < (TILE * TILE) / 256; ++i) {
        int pix = threadIdx.x + i * 256;
        int lx = pix & (TILE - 1), ly = pix >> 7;   // TILE == 128
        int gx = tx0 + lx, gy = ty0 + ly;

        float xx = fmaf((float)gx, 2.0f / 511.0f, -1.0f);
        float yy = fmaf((float)gy, 2.0f / 511.0f, -1.0f);
        float gxf = xx * t00 + yy * t01 + t02;
        float gyf = xx * t10 + yy * t11 + t12;
        float ix = (gxf + 1.0f) * 31.5f;  // 0.5*(PATCH-1)
        float iy = (gyf + 1.0f) * 31.5f;

        float x0f = floorf(ix), y0f = floorf(iy);
        float fx1 = ix - x0f,  fy1 = iy - y0f;
        float fx0 = 1.0f - fx1, fy0 = 1.0f - fy1;
        int x0 = (int)x0f, y0 = (int)y0f;
        int x1 = x0 + 1,   y1 = y0 + 1;
        bool vx0 = (x0 >= 0) & (x0 < PATCH);
        bool vx1 = (x1 >= 0) & (x1 < PATCH);
        bool vy0 = (y0 >= 0) & (y0 < PATCH);
        bool vy1 = (y1 >= 0) & (y1 < PATCH);
        float w00 = fx0 * fy0 * (float)(vx0 && vy0);
        float w10 = fx1 * fy0 * (float)(vx1 && vy0);
        float w01 = fx0 * fy1 * (float)(vx0 && vy1);
        float w11 = fx1 * fy1 * (float)(vx1 && vy1);
        int cx0 = min(max(x0, 0), PATCH - 1);
        int cx1 = min(max(x1, 0), PATCH - 1);
        int cy0 = min(max(y0, 0), PATCH - 1);
        int cy1 = min(max(y1, 0), PATCH - 1);
        int i00 = cy0 * PATCH + cx0;
        int i10 = cy0 * PATCH + cx1;
        int i01 = cy1 * PATCH + cx0;
        int i11 = cy1 * PATCH + cx1;
        size_t po = (size_t)gy * CANVAS + gx;

#pragma unroll
        for (int c = 0; c < CHANNELS; ++c) {
            const float* chp = simg + c * (PATCH * PATCH);
            const float* cha = simg + (c + CHANNELS) * (PATCH * PATCH);
            float vy_ = fmaf(chp[i00], w00, fmaf(chp[i10], w10,
                        fmaf(chp[i01], w01, chp[i11] * w11)));
            float vm_ = fmaf(cha[i00], w00, fmaf(cha[i10], w10,
                        fmaf(cha[i01], w01, cha[i11] * w11)));
            size_t co = obase + (size_t)c * CANVAS * CANVAS + po;
            __builtin_nontemporal_store(vy_, &y_out[co]);
            __builtin_nontemporal_store(vm_, &m_out[co]);
        }
    }
}

// ---------------------------------------------------------------------------
extern "C" void kernel_launch(void* const* d_in, const int* in_sizes, int n_in,
                              void* d_out, int out_size, void* d_ws, size_t ws_size,
                              hipStream_t stream) {
    (void)in_sizes; (void)n_in; (void)out_size; (void)ws_size;
    const float* b_z     = (const float*)d_in[0];
    const float* patch_w = (const float*)d_in[1];
    const float* patch_b = (const float*)d_in[2];
    const float* alpha_w = (const float*)d_in[3];
    const float* alpha_b = (const float*)d_in[4];
    const float* trans_w = (const float*)d_in[5];
    const float* trans_b = (const float*)d_in[6];
    // d_in[7], d_in[8] = canvas_h, canvas_w (== 512, hardcoded)

    float* theta = (float*)d_ws;                 // 192 floats
    float* img   = theta + 256;                  // 32*6*64*64 floats = 3 MB

    float* y_out = (float*)d_out;
    float* m_out = y_out + (size_t)BATCH * CHANNELS * CANVAS * CANVAS;

    k_theta<<<1, 192, 0, stream>>>(b_z, trans_w, trans_b, theta);

    // 1536 tiles, 8 waves (one tile each) per 256-thread block -> 192 blocks
    k_gen<<<192, 256, 0, stream>>>(b_z, patch_w, patch_b, alpha_w, alpha_b, img);

    dim3 grid(CANVAS / TILE, CANVAS / TILE, BATCH); // (4, 4, 32)
    k_render<<<grid, 256, IMG_PER_B * sizeof(float), stream>>>(img, theta, y_out, m_out);
}